// SG32_3496103379567
// MI455X (gfx1250) — compile-verified
//
#include <hip/hip_runtime.h>
#include <stdint.h>

// ---------------------------------------------------------------------------
// SGConv x31 stack on MI455X (gfx1250, wave32).
//  - One-time CSR build (count / scan / scatter) so the 31-layer loop is
//    atomic-free and streams from the 192MB L2.
//  - Aggregation: one wave per node, one lane per channel (32 ch == wave32),
//    every gathered h-row is a single 128B coalesced load.
//  - All GEMMs via v_wmma_f32_16x16x4_f32 (exact fp32, matches reference).
// ---------------------------------------------------------------------------

typedef __attribute__((ext_vector_type(2))) float v2f;
typedef __attribute__((ext_vector_type(8))) float v8f;

#define HID 32

// ---------------- CSR build ----------------

__global__ void k_zero_i32(int* p, int n) {
    for (int i = blockIdx.x * blockDim.x + threadIdx.x; i < n;
         i += gridDim.x * blockDim.x)
        p[i] = 0;
}

__global__ void k_count(const int* __restrict__ col, int* __restrict__ cnt, int E) {
    int e = blockIdx.x * blockDim.x + threadIdx.x;
    if (e < E) atomicAdd(&cnt[col[e]], 1);
}

__global__ void k_dinv(const int* __restrict__ cnt, float* __restrict__ dinv, int N) {
    int v = blockIdx.x * blockDim.x + threadIdx.x;
    if (v < N) {
        float deg = (float)(cnt[v] + 1);   // +1 self loop, always > 0
        dinv[v] = rsqrtf(deg);
    }
}

// Single-block exclusive scan over cnt[0..N) -> offs, cursor ; offs[N] = total
__global__ __launch_bounds__(1024)
void k_scan(const int* __restrict__ cnt, int* __restrict__ offs,
            int* __restrict__ cursor, int N) {
    __shared__ int sm[1024];
    __shared__ int carry;
    int t = threadIdx.x;
    if (t == 0) carry = 0;
    __syncthreads();
    for (int base = 0; base < N; base += 1024) {
        int i = base + t;
        int v = (i < N) ? cnt[i] : 0;
        sm[t] = v;
        __syncthreads();
#pragma unroll
        for (int off = 1; off < 1024; off <<= 1) {
            int add = (t >= off) ? sm[t - off] : 0;
            __syncthreads();
            sm[t] += add;
            __syncthreads();
        }
        int incl = sm[t];
        int excl = incl - v;
        if (i < N) {
            offs[i]   = carry + excl;
            cursor[i] = carry + excl;
        }
        __syncthreads();
        if (t == 1023) carry += incl;
        __syncthreads();
    }
    if (t == 0) offs[N] = carry;
}

__global__ void k_scatter(const int* __restrict__ row, const int* __restrict__ col,
                          const float* __restrict__ dinv, int* __restrict__ cursor,
                          int* __restrict__ rowS, float* __restrict__ normS, int E) {
    int e = blockIdx.x * blockDim.x + threadIdx.x;
    if (e >= E) return;
    int c = col[e], r = row[e];
    int pos = atomicAdd(&cursor[c], 1);
    rowS[pos]  = r;
    normS[pos] = dinv[r] * dinv[c];
}

// ---------------- propagation: agg[v] = dinv[v]^2*h[v] + sum norm*h[row] ----

__global__ __launch_bounds__(256)
void k_aggregate(const float* __restrict__ h, const int* __restrict__ offs,
                 const int* __restrict__ rowS, const float* __restrict__ normS,
                 const float* __restrict__ dinv, float* __restrict__ agg, int N) {
    int wave = (blockIdx.x * blockDim.x + threadIdx.x) >> 5;  // wave32: node id
    int lane = threadIdx.x & 31;                              // channel id
    if (wave >= N) return;                                    // wave-uniform
    float di  = dinv[wave];
    float acc = di * di * h[(size_t)wave * HID + lane];       // self loop
    int beg = offs[wave], end = offs[wave + 1];
    for (int base = beg; base < end; base += 32) {
        int e = base + lane;
        int u = 0; float w = 0.0f;
        if (e < end) { u = rowS[e]; w = normS[e]; }
        if (base + 32 < end) {                                // pre-touch next chunk
            __builtin_prefetch(rowS + base + 32 + lane, 0, 1);
            __builtin_prefetch(normS + base + 32 + lane, 0, 1);
        }
        int cnt = end - base; if (cnt > 32) cnt = 32;
        for (int j = 0; j < cnt; ++j) {
            int   uj = __shfl(u, j, 32);
            float wj = __shfl(w, j, 32);
            // 32 lanes read 32 consecutive floats: one 128B coalesced load
            acc = fmaf(wj, h[(size_t)uj * HID + lane], acc);
        }
    }
    agg[(size_t)wave * HID + lane] = acc;
}

// ---------------- WMMA fp32 GEMM: D = A[N,K] @ W[K,COLS] + bias (opt ReLU) --

template <int K, int COLS, bool RELU>
__global__ __launch_bounds__(256)
void k_gemm_wmma(const float* __restrict__ A, const float* __restrict__ W,
                 const float* __restrict__ bias, float* __restrict__ D, int nrows) {
    int wid  = (blockIdx.x * blockDim.x + threadIdx.x) >> 5;  // wave id
    int lane = threadIdx.x & 31;
    int tile = wid;                                           // 16 rows / wave
    if (tile * 16 >= nrows) return;                           // wave-uniform

    int m  = lane & 15;              // A: row within tile; B/D: column within block
    int kb = (lane >> 4) << 1;       // K sub-offset 0 or 2 per 16x4 fp32 layout
    int ar = tile * 16 + m;
    if (ar >= nrows) ar = nrows - 1;                          // clamp (loads only)
    const float* arow = A + (size_t)ar * K;

#pragma unroll
    for (int cb = 0; cb < COLS / 16; ++cb) {
        v8f acc = {};
#pragma unroll
        for (int k0 = 0; k0 < K; k0 += 4) {
            v2f a, b;
            a.x = arow[k0 + kb];
            a.y = arow[k0 + kb + 1];
            b.x = W[(size_t)(k0 + kb)     * COLS + cb * 16 + m];
            b.y = W[(size_t)(k0 + kb + 1) * COLS + cb * 16 + m];
            acc = __builtin_amdgcn_wmma_f32_16x16x4_f32(
                /*neg_a=*/false, a, /*neg_b=*/false, b,
                /*c_mod=*/(short)0, acc, /*reuse_a=*/false, /*reuse_b=*/false);
        }
        // D layout: VGPR r -> row (8*(lane>=16) + r), col = lane&15
        int col   = cb * 16 + m;
        float bv  = bias[col];
        int rbase = tile * 16 + ((lane >> 4) << 3);
#pragma unroll
        for (int r = 0; r < 8; ++r) {
            float v = acc[r] + bv;
            if (RELU) v = fmaxf(v, 0.0f);
            if (rbase + r < nrows)
                D[(size_t)(rbase + r) * COLS + col] = v;
        }
    }
}

// ---------------- host side ----------------

static inline char* align256(char* p) {
    return (char*)(((uintptr_t)p + 255) & ~(uintptr_t)255);
}

extern "C" void kernel_launch(void* const* d_in, const int* in_sizes, int n_in,
                              void* d_out, int out_size, void* d_ws, size_t ws_size,
                              hipStream_t stream) {
    const float* x    = (const float*)d_in[0];   // [N,128]
    const float* W0   = (const float*)d_in[1];   // [128,32]
    const float* b0   = (const float*)d_in[2];   // [32]
    const float* Ws   = (const float*)d_in[3];   // [31,32,32]
    const float* bs   = (const float*)d_in[4];   // [31,32]
    const float* W32  = (const float*)d_in[5];   // [32,64]
    const float* b32  = (const float*)d_in[6];   // [64]
    const int*   edge = (const int*)d_in[7];     // [2,E] (int32 after JAX canonicalization)
    float* out = (float*)d_out;

    const int N = in_sizes[0] / 128;
    const int E = in_sizes[7] / 2;
    const int IN_CH = 128, OUT_CH = 64, N_SG = 31;

    const int* row = edge;        // sources
    const int* col = edge + E;    // destinations

    // workspace partition
    char* p = (char*)d_ws;
    int*   cnt    = (int*)p;                 p = align256(p + sizeof(int)   * N);
    int*   offs   = (int*)p;                 p = align256(p + sizeof(int)   * (N + 1));
    int*   cursor = (int*)p;                 p = align256(p + sizeof(int)   * N);
    int*   rowS   = (int*)p;                 p = align256(p + sizeof(int)   * E);
    float* normS  = (float*)p;               p = align256(p + sizeof(float) * E);
    float* dinv   = (float*)p;               p = align256(p + sizeof(float) * N);
    float* h0     = (float*)p;               p = align256(p + sizeof(float) * (size_t)N * HID);
    float* h1     = (float*)p;               p = align256(p + sizeof(float) * (size_t)N * HID);
    float* agg    = (float*)p;               p = align256(p + sizeof(float) * (size_t)N * HID);
    (void)ws_size; (void)n_in; (void)out_size;

    const int B = 256;
    // ---- one-time CSR build (deterministic work; atomics only here) ----
    k_zero_i32<<<(N + B - 1) / B, B, 0, stream>>>(cnt, N);
    k_count<<<(E + B - 1) / B, B, 0, stream>>>(col, cnt, E);
    k_dinv<<<(N + B - 1) / B, B, 0, stream>>>(cnt, dinv, N);
    k_scan<<<1, 1024, 0, stream>>>(cnt, offs, cursor, N);
    k_scatter<<<(E + B - 1) / B, B, 0, stream>>>(row, col, dinv, cursor, rowS, normS, E);

    // ---- conv0: h = x @ W0 + b0 ----
    int tiles  = (N + 15) / 16;
    int gblk   = (tiles + (B / 32) - 1) / (B / 32);   // 8 waves / block
    k_gemm_wmma<IN_CH, HID, false><<<gblk, B, 0, stream>>>(x, W0, b0, h0, N);

    // ---- conv1..conv31: h = relu( propagate(h) @ W + b ) ----
    int ablk = (N + (B / 32) - 1) / (B / 32);         // wave per node
    float* hcur = h0;
    float* hnxt = h1;
    for (int l = 0; l < N_SG; ++l) {
        k_aggregate<<<ablk, B, 0, stream>>>(hcur, offs, rowS, normS, dinv, agg, N);
        k_gemm_wmma<HID, HID, true><<<gblk, B, 0, stream>>>(
            agg, Ws + (size_t)l * HID * HID, bs + (size_t)l * HID, hnxt, N);
        float* t = hcur; hcur = hnxt; hnxt = t;
    }

    // ---- conv32: out = h @ W32 + b32 ----
    k_gemm_wmma<HID, OUT_CH, false><<<gblk, B, 0, stream>>>(hcur, W32, b32, out, N);
}